// Boundary_CE_Loss_1932735283770
// MI455X (gfx1250) — compile-verified
//
#include <hip/hip_runtime.h>
#include <math.h>

// ---------------------------------------------------------------------------
// Boundary CE loss for MI455X (gfx1250, wave32).
//   1) edges_kernel : one-hot staging (reflect pad), separable Gaussian blur
//      as chained V_WMMA_F32_16X16X4_F32 against banded Gaussian matrices
//      (fragments precomputed branchlessly, hoisted out of the WMMA loops),
//      Sobel (edge pad), magnitude, branchless octant NMS (zero pad), double
//      threshold -> weak/strong bitmasks packed via wave32 ballot.
//   2) hyst_kernel  : one workgroup per plane; strong+weak bitmaps = 64KB LDS
//      loaded via the Tensor Data Mover (tensor_load_to_lds + s_wait_tensorcnt),
//      bit-parallel 8-neighbour dilation fixpoint with __syncthreads_or,
//      then cross-dilation -> final edge bitmask.
//   3) loss_kernel  : streaming logsumexp over 11 channels (92MB pd = the
//      roofline term, ~4us at 23.3TB/s), gather gt-class log-prob, weighted
//      reduce via wave32 shuffles + global_atomic_add_f32.
// ---------------------------------------------------------------------------

typedef float v2f __attribute__((ext_vector_type(2)));
typedef float v8f __attribute__((ext_vector_type(8)));
typedef unsigned int u32x4 __attribute__((ext_vector_type(4)));
typedef int i32x4 __attribute__((ext_vector_type(4)));
typedef int i32x8 __attribute__((ext_vector_type(8)));

#define IMG   512
#define NCLS  11
#define NBATCH 8
#define NPLANE (NBATCH * NCLS)                 /* 88 */
#define WORDS_PER_ROW 16                       /* 512 cols / 32 bits */
#define WORDS_PER_PLANE (IMG * WORDS_PER_ROW)  /* 8192 */
#define NPIX (NBATCH * IMG * IMG)              /* 2097152 == onehot.sum() */

// Normalized 5-tap Gaussian; branchless (pure v_cmp/v_cndmask).
__device__ __forceinline__ float gsel(int i) {
  float r = 0.0f;
  r = (i == 0 || i == 4) ? 0.05448868f : r;
  r = (i == 1 || i == 3) ? 0.24420135f : r;
  r = (i == 2)           ? 0.40261996f : r;
  return r;
}

__device__ __forceinline__ int reflect512(int v) {
  if (v < 0) v = -v;
  if (v > IMG - 1) v = 2 * (IMG - 1) - v;   // jnp 'reflect' (no edge repeat)
  return v;
}

// D(16x16,f32) = A(16x4,f32) * B(4x16,f32) + C   — CDNA5 WMMA, wave32.
__device__ __forceinline__ v8f wmma4(v2f a, v2f b, v8f c) {
  return __builtin_amdgcn_wmma_f32_16x16x4_f32(false, a, false, b,
                                               (short)0, c, false, false);
}

// ---------------------------------------------------------------------------
// Tensor Data Mover: 1-D load of `nwords` dwords from global -> LDS.
// D# packed per CDNA5 ISA (group0 + group1, 2-D form).
// ---------------------------------------------------------------------------
__device__ __forceinline__ void tdm_load_words(unsigned lds_off,
                                               const void* gptr,
                                               unsigned nwords) {
  unsigned long long ga = (unsigned long long)(uintptr_t)gptr;
  u32x4 g0;
  g0[0] = 1u;                                  // count=1 (valid user D#)
  g0[1] = lds_off;                             // lds_addr (bytes)
  g0[2] = (unsigned)(ga & 0xffffffffu);        // global_addr[31:0]  -> bits 95:64
  g0[3] = (unsigned)((ga >> 32) & 0x01ffffffu) // global_addr[56:32] -> bits 120:96
        | (2u << 30);                          // type=2 ("image")   -> bits 127:126
  i32x8 g1;
  g1[0] = 0x20000;                             // data_size=2 (4 bytes); mask=0
  g1[1] = (int)((nwords & 0xffffu) << 16);     // tensor_dim0[15:0]  -> bits 63:48
  g1[2] = (int)((nwords >> 16) | (1u << 16));  // tensor_dim0[31:16] | tensor_dim1=1
  g1[3] = (int)((nwords & 0xffffu) << 16);     // tile_dim0          -> bits 127:112
  g1[4] = 1;                                   // tile_dim1=1; tile_dim2=0
  g1[5] = (int)nwords;                         // tensor_dim0_stride[31:0]
  g1[6] = 0;                                   // stride hi / dim1_stride lo
  g1[7] = 0;
  i32x4 z4 = {0, 0, 0, 0};
#if defined(__clang_major__) && (__clang_major__ >= 23)
  i32x8 z8 = {0, 0, 0, 0, 0, 0, 0, 0};
  __builtin_amdgcn_tensor_load_to_lds(g0, g1, z4, z4, z8, 0);
#else
  __builtin_amdgcn_tensor_load_to_lds(g0, g1, z4, z4, 0);
#endif
}

// ---------------------------------------------------------------------------
// Kernel 1: blur (WMMA) + Sobel + NMS + thresholds -> weak/strong bitmasks
// grid = (16, 16, 88), block = 128 (4 waves)
// ---------------------------------------------------------------------------
__global__ __launch_bounds__(128)
void edges_kernel(const int* __restrict__ gt,
                  unsigned* __restrict__ weakM,
                  unsigned* __restrict__ strongM) {
  // Local frames (relative to tile origin):
  //   sO   : one-hot,   rows [-4..51], cols [-4..47]
  //   sT   : h-blurred, rows [-4..51], cols [-2..45]
  //   sB   : blurred,   rows/cols [-2..45]
  //   sMag : magnitude, rows/cols [-1..32]
  __shared__ float sO[56][56];
  __shared__ float sT[56][48];
  __shared__ float sB[48][48];
  __shared__ float sMag[34][36];

  const int origin_r = blockIdx.y * 32;
  const int origin_c = blockIdx.x * 32;
  const int plane = blockIdx.z;
  const int batch = plane / NCLS;
  const int cls   = plane % NCLS;
  const int* gtp = gt + (size_t)batch * IMG * IMG;

  // --- stage one-hot with reflect padding --------------------------------
  for (int i = threadIdx.x; i < 56 * 52; i += 128) {
    int lr = i / 52, lc = i % 52;
    int gr = reflect512(origin_r - 4 + lr);
    int gc = reflect512(origin_c - 4 + lc);
    sO[lr][lc] = (gtp[gr * IMG + gc] == cls) ? 1.0f : 0.0f;
  }

  const int lane = threadIdx.x & 31;
  const int wv   = threadIdx.x >> 5;
  const int M    = lane & 15;    // A row / B,C,D column
  const int half = lane >> 4;

  // Precompute banded-Gaussian WMMA fragments once (branchless, hoisted).
  v2f hb[5], va[5];
#pragma unroll
  for (int t = 0; t < 5; ++t) {
    hb[t].x = gsel(4 * t + half     - M);   // Gh[4t+k][n] = g[4t+k-n]
    hb[t].y = gsel(4 * t + half + 2 - M);
    va[t].x = gsel(4 * t + 2 * half     - M); // Gv[m][4t+k] = g[4t+k-m]
    va[t].y = gsel(4 * t + 2 * half + 1 - M);
  }
  __syncthreads();

  // --- horizontal blur: tmp = O * Gh, chained WMMA k=4 -------------------
  {
    int R = (wv < 3) ? wv * 16 : 36;          // row groups 0,16,32,36
    for (int C = 0; C < 3; ++C) {
      v8f acc = {};
#pragma unroll
      for (int t = 0; t < 5; ++t) {
        v2f a;
        int acol = C * 16 + 4 * t + 2 * half; // even -> 8B-aligned pair
        a.x = sO[R + M][acol];
        a.y = sO[R + M][acol + 1];
        acc = wmma4(a, hb[t], acc);
      }
#pragma unroll
      for (int v = 0; v < 8; ++v)
        sT[R + v + 8 * half][C * 16 + M] = acc[v];
    }
  }
  __syncthreads();

  // --- vertical blur: b = Gv * tmp, chained WMMA k=4 ---------------------
  for (int T = wv; T < 9; T += 4) {
    int Rb = (T / 3) * 16, Cb = (T % 3) * 16;
    v8f acc = {};
#pragma unroll
    for (int t = 0; t < 5; ++t) {
      v2f b;
      b.x = sT[Rb + 4 * t + half    ][Cb + M];
      b.y = sT[Rb + 4 * t + half + 2][Cb + M];
      acc = wmma4(va[t], b, acc);
    }
#pragma unroll
    for (int v = 0; v < 8; ++v)
      sB[Rb + v + 8 * half][Cb + M] = acc[v];
  }
  __syncthreads();

  // --- Sobel magnitude on [-1..32]^2 (edge/clamp pad at image border) ----
  const int br0 = origin_r - 2, bc0 = origin_c - 2;
  auto sampB = [&](int rr, int cc) -> float {
    rr = rr < 0 ? 0 : (rr > IMG - 1 ? IMG - 1 : rr);
    cc = cc < 0 ? 0 : (cc > IMG - 1 ? IMG - 1 : cc);
    return sB[rr - br0][cc - bc0];
  };
  for (int i = threadIdx.x; i < 34 * 34; i += 128) {
    int mr = i / 34, mc = i % 34;
    int gr = origin_r - 1 + mr, gc = origin_c - 1 + mc;
    float a00 = sampB(gr-1, gc-1), a01 = sampB(gr-1, gc), a02 = sampB(gr-1, gc+1);
    float a10 = sampB(gr,   gc-1),                         a12 = sampB(gr,   gc+1);
    float a20 = sampB(gr+1, gc-1), a21 = sampB(gr+1, gc), a22 = sampB(gr+1, gc+1);
    float gx = (a02 - a00) + 2.0f * (a12 - a10) + (a22 - a20);
    float gy = (a20 - a00) + 2.0f * (a21 - a01) + (a22 - a02);
    float m = sqrtf(gx * gx + gy * gy + 1e-6f);
    if (gr < 0 || gr > IMG - 1 || gc < 0 || gc > IMG - 1) m = 0.0f; // NMS zero pad
    sMag[mr][mc] = m;
  }
  __syncthreads();

  // --- NMS + double threshold; branchless octant; ballot-packed rows -----
  for (int rr = wv; rr < 32; rr += 4) {
    int c  = lane;
    int gr = origin_r + rr, gc = origin_c + c;
    float a00 = sampB(gr-1, gc-1), a01 = sampB(gr-1, gc), a02 = sampB(gr-1, gc+1);
    float a10 = sampB(gr,   gc-1),                         a12 = sampB(gr,   gc+1);
    float a20 = sampB(gr+1, gc-1), a21 = sampB(gr+1, gc), a22 = sampB(gr+1, gc+1);
    float gx = (a02 - a00) + 2.0f * (a12 - a10) + (a22 - a20);
    float gy = (a20 - a00) + 2.0f * (a21 - a01) + (a22 - a02);
    float mag = sMag[rr + 1][c + 1];

    // octant = mod(round(atan2(gy,gx)/45deg), 8) without atan2:
    // thresholds tan(22.5)=0.41421356, tan(67.5)=2.41421356; the <= / >=
    // choices reproduce round-half-even at the exact bin boundaries.
    float ax = fabsf(gx), ay = fabsf(gy);
    int o0 = (ay <= 0.41421356f * ax) ? 0 : ((ay >= 2.41421356f * ax) ? 2 : 1);
    int pos;
    if (gy >= 0.0f) pos = (gx >= 0.0f) ? o0 : 4 - o0;
    else            pos = (gx <  0.0f) ? o0 + 4 : ((8 - o0) & 7);
    // direction of octant `pos`; opposite octant is just the negation.
    int dr = (pos >= 5) ? 1 : ((pos >= 1 && pos <= 3) ? -1 : 0);
    int dc = (pos <= 1 || pos == 7) ? 1 : ((pos >= 3 && pos <= 5) ? -1 : 0);

    float cp = mag - sMag[rr + 1 + dr][c + 1 + dc];
    float cn = mag - sMag[rr + 1 - dr][c + 1 - dc];
    float m = (fminf(cp, cn) > 0.0f) ? mag : 0.0f;

    unsigned wk = __builtin_amdgcn_ballot_w32(m > 0.1f);  // weak-or-strong
    unsigned st = __builtin_amdgcn_ballot_w32(m > 0.2f);  // strong
    if (lane == 0) {
      size_t base = (size_t)plane * WORDS_PER_PLANE +
                    (size_t)gr * WORDS_PER_ROW + blockIdx.x;
      weakM[base]   = wk;
      strongM[base] = st;
    }
  }
}

// ---------------------------------------------------------------------------
// Kernel 2: hysteresis fixpoint + cross dilation, whole plane in 64KB LDS,
// masks DMA'd in by the Tensor Data Mover.
// grid = 88, block = 1024 (32 waves)
// ---------------------------------------------------------------------------
__global__ __launch_bounds__(1024)
void hyst_kernel(const unsigned* __restrict__ weakM,
                 const unsigned* __restrict__ strongM,
                 unsigned* __restrict__ edgeM) {
  __shared__ unsigned sS[IMG][WORDS_PER_ROW];   // 32KB strong bitmap
  __shared__ unsigned sW[IMG][WORDS_PER_ROW];   // 32KB weak bitmap
  const int plane = blockIdx.x;

  // TDM: async-load both 32KB bitmaps straight into LDS (wave 0 issues;
  // TENSORcnt is per-wave, so wave 0 also waits, then block-barrier).
  if ((threadIdx.x >> 5) == 0) {
    tdm_load_words((unsigned)(uintptr_t)&sS[0][0],
                   strongM + (size_t)plane * WORDS_PER_PLANE, WORDS_PER_PLANE);
    tdm_load_words((unsigned)(uintptr_t)&sW[0][0],
                   weakM + (size_t)plane * WORDS_PER_PLANE, WORDS_PER_PLANE);
    __builtin_amdgcn_s_wait_tensorcnt(0);
  }
  __syncthreads();

  // Jacobi fixpoint: S |= W & dilate8(S); stop when no bit flips.
  for (int iter = 0; iter < 512; ++iter) {
    unsigned ns[8];
    unsigned ch = 0;
#pragma unroll
    for (int k = 0; k < 8; ++k) {
      int i = threadIdx.x + k * 1024;
      int r = i >> 4, wc = i & 15;
      unsigned dil = 0;
#pragma unroll
      for (int dr = -1; dr <= 1; ++dr) {
        int rr = r + dr;
        if (rr < 0 || rr > IMG - 1) continue;       // RING conv zero pad
        unsigned sc = sS[rr][wc];
        unsigned sl = (wc > 0)  ? sS[rr][wc - 1] : 0u;
        unsigned sr = (wc < 15) ? sS[rr][wc + 1] : 0u;
        dil |= sc | (sc << 1) | (sl >> 31) | (sc >> 1) | (sr << 31);
      }
      unsigned s0 = sS[r][wc];
      unsigned nv = s0 | (sW[r][wc] & dil);
      ns[k] = nv;
      ch |= nv ^ s0;
    }
    __syncthreads();
#pragma unroll
    for (int k = 0; k < 8; ++k) {
      int i = threadIdx.x + k * 1024;
      sS[i >> 4][i & 15] = ns[k];
    }
    int any = __syncthreads_or(ch != 0u);
    if (!any) break;
  }

  // dilate_cross (zero pad) of converged strong set -> final edge mask
  unsigned* ep = edgeM + (size_t)plane * WORDS_PER_PLANE;
  for (int i = threadIdx.x; i < WORDS_PER_PLANE; i += 1024) {
    int r = i >> 4, wc = i & 15;
    unsigned c  = sS[r][wc];
    unsigned l  = (wc > 0)  ? sS[r][wc - 1] : 0u;
    unsigned rg = (wc < 15) ? sS[r][wc + 1] : 0u;
    unsigned up = (r > 0)       ? sS[r - 1][wc] : 0u;
    unsigned dn = (r < IMG - 1) ? sS[r + 1][wc] : 0u;
    ep[i] = c | (c << 1) | (l >> 31) | (c >> 1) | (rg << 31) | up | dn;
  }
}

// ---------------------------------------------------------------------------
// Kernel 3: log_softmax gather + weighted reduction (memory bound: 92MB pd)
// ---------------------------------------------------------------------------
__global__ __launch_bounds__(256)
void loss_kernel(const float* __restrict__ pd,
                 const int* __restrict__ gt,
                 const float* __restrict__ wgt,
                 const unsigned* __restrict__ edgeM,
                 float* __restrict__ out) {
  int idx = blockIdx.x * blockDim.x + threadIdx.x;
  float v = 0.0f;
  if (idx < NPIX) {
    int b   = idx >> 18;
    int pix = idx & (IMG * IMG - 1);
    const float* p = pd + (size_t)b * NCLS * IMG * IMG + pix;
    __builtin_prefetch(p + (size_t)5 * IMG * IMG, 0, 0);  // global_prefetch_b8
    float x[NCLS];
    float mx = -INFINITY;
#pragma unroll
    for (int c = 0; c < NCLS; ++c) {
      x[c] = p[(size_t)c * IMG * IMG];
      mx = fmaxf(mx, x[c]);
    }
    float s = 0.0f;
#pragma unroll
    for (int c = 0; c < NCLS; ++c)
      s += expf(x[c] - mx);
    float lse = mx + logf(s);
    int c = gt[idx];
    float xc = p[(size_t)c * IMG * IMG];   // re-load: keeps x[] constant-indexed
    int r = pix >> 9, col = pix & (IMG - 1);
    unsigned wbits = edgeM[(size_t)(b * NCLS + c) * WORDS_PER_PLANE +
                           (size_t)r * WORDS_PER_ROW + (col >> 5)];
    float edge = ((wbits >> (col & 31)) & 1u) ? 1.0f : 0.0f;
    v = -(wgt[c] * (xc - lse) * (1.0f + 2.0f * edge)) * (1.0f / (float)NPIX);
  }
  // wave32 reduction, one atomic per wave
#pragma unroll
  for (int off = 16; off > 0; off >>= 1)
    v += __shfl_down(v, off, 32);
  if ((threadIdx.x & 31) == 0)
    atomicAdd(out, v);
}

__global__ void zero_out_kernel(float* out) { out[0] = 0.0f; }

// ---------------------------------------------------------------------------
extern "C" void kernel_launch(void* const* d_in, const int* in_sizes, int n_in,
                              void* d_out, int out_size, void* d_ws, size_t ws_size,
                              hipStream_t stream) {
  (void)in_sizes; (void)n_in; (void)out_size; (void)ws_size;
  const float* pd  = (const float*)d_in[0];
  const int*   gt  = (const int*)d_in[1];
  const float* wgt = (const float*)d_in[2];
  float* out = (float*)d_out;

  unsigned* weakM   = (unsigned*)d_ws;                         // 2.88MB each
  unsigned* strongM = weakM   + (size_t)NPLANE * WORDS_PER_PLANE;
  unsigned* edgeM   = strongM + (size_t)NPLANE * WORDS_PER_PLANE;

  zero_out_kernel<<<1, 1, 0, stream>>>(out);

  dim3 eg(IMG / 32, IMG / 32, NPLANE);                         // 16 x 16 x 88
  edges_kernel<<<eg, 128, 0, stream>>>(gt, weakM, strongM);

  hyst_kernel<<<NPLANE, 1024, 0, stream>>>(weakM, strongM, edgeM);

  loss_kernel<<<NPIX / 256, 256, 0, stream>>>(pd, gt, wgt, edgeM, out);
}